// PantopicQuality_90417651516102
// MI455X (gfx1250) — compile-verified
//
#include <hip/hip_runtime.h>

#define L_LAB 128
#define MATN  (L_LAB * L_LAB)   // 16384 confusion-matrix cells (64 KB as u32)
#define HIST_BLOCK 512

#if defined(__gfx1250__) && __has_builtin(__builtin_amdgcn_global_store_async_from_lds_b128)
#define USE_ASYNC_FLUSH 1
#else
#define USE_ASYNC_FLUSH 0
#endif

#if USE_ASYNC_FLUSH
// The toolchain's builtin signature (from the round-1 diagnostic) takes
// 16-byte int vectors: (v4i __device__*, v4i __shared__*, imm offset, imm cpol)
typedef int pq_v4i __attribute__((vector_size(16)));
typedef __attribute__((address_space(1))) pq_v4i* pq_gptr;
typedef __attribute__((address_space(3))) pq_v4i* pq_lptr;
#endif

// ---------------------------------------------------------------------------
// Kernel 1: per-workgroup private 128x128 histogram in LDS (ds_add_u32),
// flushed to a private workspace slot via CDNA5 async LDS->global stores.
// ---------------------------------------------------------------------------
__global__ __launch_bounds__(HIST_BLOCK)
void pq_hist_partial(const int* __restrict__ gt, const int* __restrict__ pred,
                     unsigned* __restrict__ partials, int n4)
{
    __shared__ unsigned hist[MATN];
    const int t = threadIdx.x;
    for (int i = t; i < MATN; i += HIST_BLOCK) hist[i] = 0u;
    __syncthreads();

    const int4* __restrict__ g4 = (const int4*)gt;
    const int4* __restrict__ p4 = (const int4*)pred;
    const int stride = gridDim.x * HIST_BLOCK;
    for (int i = blockIdx.x * HIST_BLOCK + t; i < n4; i += stride) {
        // prefetch next grid-stride tile into L2 (global_prefetch_b8)
        __builtin_prefetch((const void*)(g4 + i + stride), 0, 1);
        __builtin_prefetch((const void*)(p4 + i + stride), 0, 1);
        int4 g = g4[i];
        int4 p = p4[i];
        atomicAdd(&hist[(g.x << 7) + p.x], 1u);
        atomicAdd(&hist[(g.y << 7) + p.y], 1u);
        atomicAdd(&hist[(g.z << 7) + p.z], 1u);
        atomicAdd(&hist[(g.w << 7) + p.w], 1u);
    }
    __syncthreads();

    unsigned* __restrict__ dst = partials + (size_t)blockIdx.x * MATN;
#if USE_ASYNC_FLUSH
    // 16 B per lane per op, straight LDS -> L2, tracked by ASYNCcnt.
    for (int i = 4 * t; i < MATN; i += 4 * HIST_BLOCK) {
        __builtin_amdgcn_global_store_async_from_lds_b128(
            (pq_gptr)(dst + i), (pq_lptr)(&hist[i]), 0, 0);
    }
#if __has_builtin(__builtin_amdgcn_s_wait_asynccnt)
    __builtin_amdgcn_s_wait_asynccnt(0);
#else
    asm volatile("s_wait_asynccnt 0" ::: "memory");
#endif
#else
    for (int i = t; i < MATN; i += HIST_BLOCK) dst[i] = hist[i];
#endif
}

// ---------------------------------------------------------------------------
// Kernel 2: sum the per-workgroup partial matrices into the final matrix.
// ---------------------------------------------------------------------------
__global__ __launch_bounds__(256)
void pq_reduce(const unsigned* __restrict__ partials, unsigned* __restrict__ mat,
               int nwg)
{
    const int cell = blockIdx.x * 256 + threadIdx.x;   // grid is exactly MATN/256
    unsigned s = 0u;
    for (int w = 0; w < nwg; ++w)
        s += partials[(size_t)w * MATN + cell];
    mat[cell] = s;
}

// ---------------------------------------------------------------------------
// Kernel 3: single wave32 performs the sequential PQ matching.
// Lane l owns pred labels {l, l+32, l+64, l+96}. The inner p-loop of the
// reference is parallel within one g iteration (matched_pr only feeds the
// p==0 merged set of *later* g iterations), so each g is a few shfl_xor
// butterflies.
// ---------------------------------------------------------------------------
__global__ __launch_bounds__(32)
void pq_finalize(const unsigned* __restrict__ M, float* __restrict__ out, float n_pix)
{
    const int lane = threadIdx.x;

    // area_pr[p] (column sums) for this lane's 4 pred labels
    float apr[4] = {0.f, 0.f, 0.f, 0.f};
    for (int g = 0; g < L_LAB; ++g)
        for (int j = 0; j < 4; ++j)
            apr[j] += (float)M[g * L_LAB + lane + 32 * j];

    const float area_pr0 = __shfl(apr[0], 0, 32);
    bool prp[4], mpr[4];
    for (int j = 0; j < 4; ++j) { prp[j] = apr[j] > 0.f; mpr[j] = false; }

    float S_area = 0.f;        // sum of area_pr over matched positive preds
    float last_iou = 0.f;      // the reference's leaked loop variable
    int   tp = 0;
    bool  any_pos_gt_matched = false;
    float unmatched_gt = 0.f;
    float area_gt0 = 0.f;

    for (int g = 0; g < L_LAB; ++g) {
        float r[4];
        float a_g = 0.f, mrow = 0.f;   // row sum; row sum over merged preds
        for (int j = 0; j < 4; ++j) {
            r[j] = (float)M[g * L_LAB + lane + 32 * j];
            a_g += r[j];
            if (mpr[j] && (lane + 32 * j) >= 1) mrow += r[j];
        }
        for (int m = 16; m >= 1; m >>= 1) {
            a_g  += __shfl_xor(a_g, m, 32);
            mrow += __shfl_xor(mrow, m, 32);
        }
        if (g == 0) area_gt0 = a_g;
        const bool  g_ok = a_g > 0.f;
        const float r0   = __shfl(r[0], 0, 32);

        // ---- p == 0 (identical on every lane) ----
        const float inter0   = r0 + mrow;
        const float ap0      = area_pr0 + S_area;
        const bool  present0 = (((n_pix - a_g) > 0.f) || (inter0 > 0.f)) && g_ok;
        const float iou0     = inter0 / fmaxf(a_g + ap0 - inter0, 1.0f);
        const bool  match0   = present0 && (iou0 > 0.5f);

        // ---- positive p, lane-parallel ----
        int   bp = present0 ? 0 : -1;   // largest p with present==true
        float bi = iou0;                // its iou
        int   nmatch = 0, anym = 0;
        float narea = 0.f;
        for (int j = 0; j < 4; ++j) {
            const int p = lane + 32 * j;
            if (p == 0) continue;
            const bool  present = prp[j] && !mpr[j] && (r[j] > 0.f) && g_ok;
            const float iou     = r[j] / fmaxf(a_g + apr[j] - r[j], 1.0f);
            const bool  match   = present && (iou > 0.5f);
            if (present && p > bp) { bp = p; bi = iou; }
            if (match) { nmatch++; narea += apr[j]; mpr[j] = true; anym = 1; }
        }
        for (int m = 16; m >= 1; m >>= 1) {
            nmatch += __shfl_xor(nmatch, m, 32);
            narea  += __shfl_xor(narea, m, 32);
            anym   |= __shfl_xor(anym, m, 32);
            const int   obp = __shfl_xor(bp, m, 32);
            const float obi = __shfl_xor(bi, m, 32);
            if (obp > bp) { bp = obp; bi = obi; }
        }
        tp     += nmatch + (match0 ? 1 : 0);
        S_area += narea;
        if (bp >= 0) last_iou = bi;     // last present pair in (g,p) order
        const bool matched_g = match0 || (anym != 0);
        if (g >= 1) {
            if (matched_g) any_pos_gt_matched = true;
            if (g_ok && !matched_g) unmatched_gt += 1.f;
        }
    }

    const float zero_gt = (area_gt0 > 0.f || any_pos_gt_matched) ? 1.f : 0.f;
    const float fn = zero_gt + unmatched_gt;

    float unmatched_pr = 0.f;
    int   any_pos_pr = 0;
    for (int j = 0; j < 4; ++j) {
        const int p = lane + 32 * j;
        if (p >= 1) {
            if (mpr[j]) any_pos_pr = 1;
            if (prp[j] && !mpr[j]) unmatched_pr += 1.f;
        }
    }
    for (int m = 16; m >= 1; m >>= 1) {
        unmatched_pr += __shfl_xor(unmatched_pr, m, 32);
        any_pos_pr   |= __shfl_xor(any_pos_pr, m, 32);
    }
    const float zero_pr = (area_pr0 > 0.f || any_pos_pr) ? 1.f : 0.f;
    const float fp = zero_pr + unmatched_pr;

    if (lane == 0)
        out[0] = last_iou / ((float)tp + 0.5f * (fn + fp));
}

// ---------------------------------------------------------------------------
extern "C" void kernel_launch(void* const* d_in, const int* in_sizes, int n_in,
                              void* d_out, int out_size, void* d_ws, size_t ws_size,
                              hipStream_t stream)
{
    const int* gt   = (const int*)d_in[0];
    const int* pred = (const int*)d_in[1];
    float* out = (float*)d_out;
    const int npix = in_sizes[0];
    const int n4   = npix >> 2;          // 4096*4096 is divisible by 4

    unsigned* mat      = (unsigned*)d_ws;    // final matrix: 64 KB
    unsigned* partials = mat + MATN;         // NWG private 64 KB slots

    // size grid from workspace: need (nwg + 1) * 64 KB
    long slots = (long)(ws_size / (MATN * sizeof(unsigned)));
    int nwg = (int)(slots - 1);
    if (nwg < 1)   nwg = 1;
    if (nwg > 512) nwg = 512;

    pq_hist_partial<<<nwg, HIST_BLOCK, 0, stream>>>(gt, pred, partials, n4);
    pq_reduce<<<MATN / 256, 256, 0, stream>>>(partials, mat, nwg);
    pq_finalize<<<1, 32, 0, stream>>>(mat, out, (float)npix);
}